// AdjMlp_18854906429731
// MI455X (gfx1250) — compile-verified
//
#include <hip/hip_runtime.h>

#define OUT_F    256
#define BUCKET_K 24
#define OVF_MAX  4096

#if __has_builtin(__builtin_amdgcn_global_load_async_to_lds_b128)
#define HAVE_ASYNC_LDS 1
#else
#define HAVE_ASYNC_LDS 0
#endif

#define GLOBAL_AS __attribute__((address_space(1)))
#define LDS_AS    __attribute__((address_space(3)))

// Exact type the builtin expects (per hipcc diagnostic): vector_size(16) int
typedef int v4i __attribute__((vector_size(4 * sizeof(int))));

// ---------------------------------------------------------------------------
// Pass 1: zero per-row counters + overflow counter
// ---------------------------------------------------------------------------
__global__ void init_counts_kernel(int* __restrict__ counts, int* __restrict__ ovf_cnt,
                                   int size) {
    int i = blockIdx.x * blockDim.x + threadIdx.x;
    if (i < size) counts[i] = 0;
    if (i == 0) *ovf_cnt = 0;
}

// ---------------------------------------------------------------------------
// Pass 2: bucket edge ids by destination row (counting scatter)
// ---------------------------------------------------------------------------
__global__ void fill_buckets_kernel(const int* __restrict__ rows,
                                    int* __restrict__ counts,
                                    int* __restrict__ bucket,
                                    int* __restrict__ ovf_cnt,
                                    int* __restrict__ ovf_list,
                                    int nnz) {
    int e = blockIdx.x * blockDim.x + threadIdx.x;
    if (e >= nnz) return;
    int r = rows[e];
    int slot = atomicAdd(&counts[r], 1);
    if (slot < BUCKET_K) {
        bucket[(size_t)r * BUCKET_K + slot] = e;
    } else {
        int o = atomicAdd(ovf_cnt, 1);
        if (o < OVF_MAX) ovf_list[o] = e;
    }
}

// ---------------------------------------------------------------------------
// Pass 3: one wave32 per output row. Edges of the row are visited in
// ascending edge-id order (deterministic FP sum). Each 1KB weight row is
// staged into a per-wave LDS buffer via async global->LDS b128 copies
// (CDNA5 ASYNCcnt path), then accumulated 8 floats/lane and the output row
// is written exactly once with two b128 stores (no atomics, no zero-init).
// ---------------------------------------------------------------------------
__global__ __launch_bounds__(256) void gather_rows_kernel(
        const int*   __restrict__ cols,
        const int*   __restrict__ counts,
        const int*   __restrict__ bucket,
        const float* __restrict__ weight,
        float*       __restrict__ out,
        int size) {
#if HAVE_ASYNC_LDS
    __shared__ float smem[8 * OUT_F];   // 8 waves/block * 1KB
#endif
    const int wave = threadIdx.x >> 5;
    const int lane = threadIdx.x & 31;
    const int row  = blockIdx.x * 8 + wave;
    if (row >= size) return;

    int n = counts[row];
    if (n > BUCKET_K) n = BUCKET_K;

    float4 acc0 = make_float4(0.f, 0.f, 0.f, 0.f);
    float4 acc1 = make_float4(0.f, 0.f, 0.f, 0.f);

    const int* brow = bucket + (size_t)row * BUCKET_K;
    int last = -1;
    for (int s = 0; s < n; ++s) {
        // selection of next-smallest edge id (n is tiny, avg 1)
        int e = 0x7fffffff;
        for (int i = 0; i < n; ++i) {
            int v = brow[i];
            if (v > last && v < e) e = v;
        }
        last = e;
        const float* wrow = weight + (size_t)cols[e] * OUT_F;
#if HAVE_ASYNC_LDS
        float* lbuf = smem + wave * OUT_F;
        // copy 1KB row: 2 x (32 lanes x 16B) async global->LDS
        {
            v4i* g0 = (v4i*)(wrow + lane * 4);
            v4i* g1 = (v4i*)(wrow + 128 + lane * 4);
            v4i* l0 = (v4i*)(lbuf + lane * 4);
            v4i* l1 = (v4i*)(lbuf + 128 + lane * 4);
            __builtin_amdgcn_global_load_async_to_lds_b128(
                (GLOBAL_AS v4i*)g0, (LDS_AS v4i*)l0, 0, 0);
            __builtin_amdgcn_global_load_async_to_lds_b128(
                (GLOBAL_AS v4i*)g1, (LDS_AS v4i*)l1, 0, 0);
        }
        asm volatile("s_wait_asynccnt 0" ::: "memory");
        const float4 a = *(const float4*)(lbuf + lane * 8);
        const float4 b = *(const float4*)(lbuf + lane * 8 + 4);
#else
        const float4 a = *(const float4*)(wrow + lane * 8);
        const float4 b = *(const float4*)(wrow + lane * 8 + 4);
#endif
        acc0.x += a.x; acc0.y += a.y; acc0.z += a.z; acc0.w += a.w;
        acc1.x += b.x; acc1.y += b.y; acc1.z += b.z; acc1.w += b.w;
    }

    float* orow = out + (size_t)row * OUT_F + lane * 8;
    *(float4*)(orow)     = acc0;
    *(float4*)(orow + 4) = acc1;
}

// ---------------------------------------------------------------------------
// Pass 4: astronomically-rare bucket overflow, handled correctly with
// fp32 global atomics AFTER the row stores.
// ---------------------------------------------------------------------------
__global__ void overflow_kernel(const int* __restrict__ rows,
                                const int* __restrict__ cols,
                                const float* __restrict__ weight,
                                float* __restrict__ out,
                                const int* __restrict__ ovf_cnt,
                                const int* __restrict__ ovf_list) {
    int nov = *ovf_cnt;
    if (nov > OVF_MAX) nov = OVF_MAX;
    int gid  = blockIdx.x * blockDim.x + threadIdx.x;
    int wave = gid >> 5;
    int lane = gid & 31;
    int nwaves = (gridDim.x * blockDim.x) >> 5;
    for (int i = wave; i < nov; i += nwaves) {
        int e = ovf_list[i];
        const float* wrow = weight + (size_t)cols[e] * OUT_F;
        float* orow = out + (size_t)rows[e] * OUT_F;
#pragma unroll
        for (int j = 0; j < 8; ++j)
            atomicAdd(orow + lane + j * 32, wrow[lane + j * 32]);
    }
}

// ---------------------------------------------------------------------------
// Fallback path if workspace is too small: zero-init + atomic scatter
// ---------------------------------------------------------------------------
__global__ void zero_out_kernel(float4* __restrict__ out, long n4) {
    long i = (long)blockIdx.x * blockDim.x + threadIdx.x;
    if (i < n4) out[i] = make_float4(0.f, 0.f, 0.f, 0.f);
}

__global__ void scatter_atomic_kernel(const int* __restrict__ rows,
                                      const int* __restrict__ cols,
                                      const float* __restrict__ weight,
                                      float* __restrict__ out, int nnz) {
    int gid  = blockIdx.x * blockDim.x + threadIdx.x;
    int wave = gid >> 5;
    int lane = gid & 31;
    if (wave >= nnz) return;
    const float* wrow = weight + (size_t)cols[wave] * OUT_F;
    float* orow = out + (size_t)rows[wave] * OUT_F;
#pragma unroll
    for (int j = 0; j < 8; ++j)
        atomicAdd(orow + lane + j * 32, wrow[lane + j * 32]);
}

// ---------------------------------------------------------------------------
extern "C" void kernel_launch(void* const* d_in, const int* in_sizes, int n_in,
                              void* d_out, int out_size, void* d_ws, size_t ws_size,
                              hipStream_t stream) {
    (void)n_in;
    const int nnz = in_sizes[0] / 2;               // adj is [2, nnz]
    const int* rows = (const int*)d_in[0];
    const int* cols = rows + nnz;
    // d_in[1] is the python-int "size" (on device, unused: derive from host sizes)
    const float* weight = (const float*)d_in[2];
    float* out = (float*)d_out;
    const int size = out_size / OUT_F;

    int* counts   = (int*)d_ws;
    int* ovf_cnt  = counts + size;
    int* ovf_list = ovf_cnt + 64;
    int* bucket   = ovf_list + OVF_MAX;
    size_t need = ((size_t)size * (1 + BUCKET_K) + 64 + OVF_MAX) * sizeof(int);

    if (ws_size >= need) {
        int b = (size + 255) / 256;
        init_counts_kernel<<<b, 256, 0, stream>>>(counts, ovf_cnt, size);
        b = (nnz + 255) / 256;
        fill_buckets_kernel<<<b, 256, 0, stream>>>(rows, counts, bucket,
                                                   ovf_cnt, ovf_list, nnz);
        b = (size + 7) / 8;                         // one wave32 per row
        gather_rows_kernel<<<b, 256, 0, stream>>>(cols, counts, bucket,
                                                  weight, out, size);
        overflow_kernel<<<16, 256, 0, stream>>>(rows, cols, weight, out,
                                                ovf_cnt, ovf_list);
    } else {
        long n4 = (long)out_size / 4;
        int b = (int)((n4 + 255) / 256);
        zero_out_kernel<<<b, 256, 0, stream>>>((float4*)out, n4);
        b = (nnz * 32 + 255) / 256;                 // one wave32 per edge
        scatter_atomic_kernel<<<b, 256, 0, stream>>>(rows, cols, weight, out, nnz);
    }
}